// GraphSAGE_43671227466485
// MI455X (gfx1250) — compile-verified
//
#include <hip/hip_runtime.h>

typedef float v2f __attribute__((ext_vector_type(2)));
typedef float v8f __attribute__((ext_vector_type(8)));

#define N_NODES 50000
#define N_EDGES 800000
#define F 128          // IN_FEATS == H_FEATS
#define NC 16          // NUM_CLASSES

// ---------------------------------------------------------------- zero fill
__global__ void zero_f32(float* __restrict__ p, long n) {
  long i = (long)blockIdx.x * blockDim.x + threadIdx.x;
  long stride = (long)gridDim.x * blockDim.x;
  for (; i < n; i += stride) p[i] = 0.0f;
}

// ------------------------------------------------- edge scatter (segment sum)
// One wave32 per edge: 32 lanes x float4 == 128 features.
// unsafeAtomicAdd -> hardware GLOBAL_ATOMIC_ADD_F32 (L2-resident target).
__global__ void edge_scatter128(const float* __restrict__ feat,
                                const int* __restrict__ src,
                                const int* __restrict__ dst,
                                float* __restrict__ sum,
                                float* __restrict__ deg,
                                int n_edges) {
  int wave = (int)((blockIdx.x * (long)blockDim.x + threadIdx.x) >> 5);
  int lane = threadIdx.x & 31;
  if (wave >= n_edges) return;
  int s = src[wave];
  int d = dst[wave];
  const float4* fp = (const float4*)(feat + (long)s * F);
  float4 v = fp[lane];
  float* op = sum + (long)d * F + lane * 4;
  unsafeAtomicAdd(op + 0, v.x);
  unsafeAtomicAdd(op + 1, v.y);
  unsafeAtomicAdd(op + 2, v.z);
  unsafeAtomicAdd(op + 3, v.w);
  if (deg != nullptr && lane == 0) unsafeAtomicAdd(deg + d, 1.0f);
}

// --------------------------------------------------- deg -> 1/max(deg,1)
__global__ void finalize_invdeg(float* __restrict__ deg, int n) {
  int i = blockIdx.x * blockDim.x + threadIdx.x;
  if (i < n) deg[i] = 1.0f / fmaxf(deg[i], 1.0f);
}

// ------------------------------------------ layer 1: dual-GEMM + mean + ReLU
// H = relu(X @ Wself + diag(invdeg) * (S @ Wneigh) + bias)
// Per-wave 16x16 tile, two independent v_wmma_f32_16x16x4_f32 chains.
__global__ void sage_layer1(const float* __restrict__ X,
                            const float* __restrict__ S,
                            const float* __restrict__ invdeg,
                            const float* __restrict__ Wself,
                            const float* __restrict__ Wneigh,
                            const float* __restrict__ bias,
                            float* __restrict__ H) {
  const int col_tiles = F / 16;  // 8
  int tile = blockIdx.x * (blockDim.x >> 5) + (threadIdx.x >> 5);
  if (tile >= (N_NODES / 16) * col_tiles) return;
  int rt = tile / col_tiles;
  int ct = tile - rt * col_tiles;
  int row0 = rt * 16, col0 = ct * 16;
  int lane = threadIdx.x & 31;
  int half = lane >> 4;      // K-half for A/B fragments
  int r    = lane & 15;      // row (A) / col (B,C,D) within tile

  const float* xrow = X + (long)(row0 + r) * F + 2 * half;
  const float* srow = S + (long)(row0 + r) * F + 2 * half;

  v8f acc_s = {};
  v8f acc_n = {};
#pragma unroll
  for (int k0 = 0; k0 < F; k0 += 4) {
    v2f a_s = { xrow[k0], xrow[k0 + 1] };
    v2f a_n = { srow[k0], srow[k0 + 1] };
    int kb = k0 + 2 * half;
    v2f b_s = { Wself[(long)kb * F + col0 + r],  Wself[(long)(kb + 1) * F + col0 + r] };
    v2f b_n = { Wneigh[(long)kb * F + col0 + r], Wneigh[(long)(kb + 1) * F + col0 + r] };
    acc_s = __builtin_amdgcn_wmma_f32_16x16x4_f32(false, a_s, false, b_s,
                                                  (short)0, acc_s, false, false);
    acc_n = __builtin_amdgcn_wmma_f32_16x16x4_f32(false, a_n, false, b_n,
                                                  (short)0, acc_n, false, false);
  }

  float bv = bias[col0 + r];
#pragma unroll
  for (int v = 0; v < 8; v++) {
    int row = row0 + v + 8 * half;
    float val = acc_s[v] + invdeg[row] * acc_n[v] + bv;
    H[(long)row * F + col0 + r] = fmaxf(val, 0.0f);
  }
}

// --------------------------------------- layer 2: dual-GEMM + mean (N = 16)
__global__ void sage_layer2(const float* __restrict__ Hin,
                            const float* __restrict__ S,
                            const float* __restrict__ invdeg,
                            const float* __restrict__ Wself,   // [128,16]
                            const float* __restrict__ Wneigh,  // [128,16]
                            const float* __restrict__ bias,    // [16]
                            float* __restrict__ O) {           // [N,16]
  int tile = blockIdx.x * (blockDim.x >> 5) + (threadIdx.x >> 5);
  if (tile >= N_NODES / 16) return;
  int row0 = tile * 16;
  int lane = threadIdx.x & 31;
  int half = lane >> 4;
  int r    = lane & 15;

  const float* hrow = Hin + (long)(row0 + r) * F + 2 * half;
  const float* srow = S + (long)(row0 + r) * F + 2 * half;

  v8f acc_s = {};
  v8f acc_n = {};
#pragma unroll
  for (int k0 = 0; k0 < F; k0 += 4) {
    v2f a_s = { hrow[k0], hrow[k0 + 1] };
    v2f a_n = { srow[k0], srow[k0 + 1] };
    int kb = k0 + 2 * half;
    v2f b_s = { Wself[(long)kb * NC + r],  Wself[(long)(kb + 1) * NC + r] };
    v2f b_n = { Wneigh[(long)kb * NC + r], Wneigh[(long)(kb + 1) * NC + r] };
    acc_s = __builtin_amdgcn_wmma_f32_16x16x4_f32(false, a_s, false, b_s,
                                                  (short)0, acc_s, false, false);
    acc_n = __builtin_amdgcn_wmma_f32_16x16x4_f32(false, a_n, false, b_n,
                                                  (short)0, acc_n, false, false);
  }

  float bv = bias[r];
#pragma unroll
  for (int v = 0; v < 8; v++) {
    int row = row0 + v + 8 * half;
    O[(long)row * NC + r] = acc_s[v] + invdeg[row] * acc_n[v] + bv;
  }
}

// ---------------------------------------------------------------- launcher
extern "C" void kernel_launch(void* const* d_in, const int* in_sizes, int n_in,
                              void* d_out, int out_size, void* d_ws, size_t ws_size,
                              hipStream_t stream) {
  const float* in_feat = (const float*)d_in[0];
  const int*   src     = (const int*)d_in[1];
  const int*   dst     = (const int*)d_in[2];
  const float* Ws1     = (const float*)d_in[3];
  const float* Wn1     = (const float*)d_in[4];
  const float* b1      = (const float*)d_in[5];
  const float* Ws2     = (const float*)d_in[6];
  const float* Wn2     = (const float*)d_in[7];
  const float* b2      = (const float*)d_in[8];
  float* out = (float*)d_out;

  // workspace layout: sum [N,128] | h1 [N,128] | deg/invdeg [N]
  float* sum = (float*)d_ws;
  float* h1  = sum + (size_t)N_NODES * F;
  float* deg = h1 + (size_t)N_NODES * F;

  const long NF = (long)N_NODES * F;

  // layer 1 aggregation
  zero_f32<<<2048, 256, 0, stream>>>(sum, NF);
  zero_f32<<<64, 256, 0, stream>>>(deg, N_NODES);
  edge_scatter128<<<(N_EDGES * 32) / 256, 256, 0, stream>>>(in_feat, src, dst,
                                                            sum, deg, N_EDGES);
  finalize_invdeg<<<(N_NODES + 255) / 256, 256, 0, stream>>>(deg, N_NODES);

  // layer 1 fused GEMMs: 3125 row tiles * 8 col tiles, 8 waves/block
  sage_layer1<<<(N_NODES / 16) * (F / 16) / 8, 256, 0, stream>>>(
      in_feat, sum, deg, Ws1, Wn1, b1, h1);

  // layer 2 aggregation (reuse sum buffer, same invdeg)
  zero_f32<<<2048, 256, 0, stream>>>(sum, NF);
  edge_scatter128<<<(N_EDGES * 32) / 256, 256, 0, stream>>>(h1, src, dst,
                                                            sum, nullptr, N_EDGES);
  // layer 2 fused GEMMs: 3125 row tiles, 8 waves/block
  sage_layer2<<<(N_NODES / 16 + 7) / 8, 256, 0, stream>>>(
      h1, sum, deg, Ws2, Wn2, b2, out);
}